// LJLKWholePoseScoringModule_4526895530581
// MI455X (gfx1250) — compile-verified
//
#include <hip/hip_runtime.h>
#include <hip/hip_bf16.h>
#include <math.h>

#define P_POSE 8
#define N_ATOM 2048
#define T_TYPE 64
#define NT (N_ATOM / 16)   // 128 j/i tiles

typedef __attribute__((ext_vector_type(2))) float v2f;
typedef __attribute__((ext_vector_type(8))) float v8f;

__device__ __forceinline__ float wave_reduce_add(float v) {
#pragma unroll
    for (int off = 16; off > 0; off >>= 1)
        v += __shfl_xor(v, off, 32);
    return v;
}

__global__ void zero_out_kernel(float* out, int n) {
    int i = threadIdx.x;
    if (i < n) out[i] = 0.0f;
}

__launch_bounds__(256)
__global__ void ljlk_pose_kernel(const float* __restrict__ coords,
                                 const float* __restrict__ ljlk_params,
                                 const int*   __restrict__ atom_types,
                                 const int*   __restrict__ path_dist,
                                 float*       __restrict__ out) {
    __shared__ float tp[T_TYPE][5];   // transformed per-type params
    __shared__ float rowp[16][5];     // per-row-atom params for this i-tile

    const int tid  = threadIdx.x;
    const int wave = tid >> 5;
    const int lane = tid & 31;
    const int half = lane >> 4;       // 0: lanes 0-15, 1: lanes 16-31
    const int col  = lane & 15;       // column / row-within-tile index

    const int p     = blockIdx.x / NT;
    const int ti    = blockIdx.x % NT;
    const int iBase = ti * 16;

    // ---- phase 1: type table (physical-range mapping + derived values) ----
    if (tid < T_TYPE) {
        float p0 = ljlk_params[tid * 5 + 0];
        float p1 = ljlk_params[tid * 5 + 1];
        float p2 = ljlk_params[tid * 5 + 2];
        float p3 = ljlk_params[tid * 5 + 3];
        float p4 = ljlk_params[tid * 5 + 4];
        float lam = 3.0f + 0.5f * p3;
        tp[tid][0] = 1.0f + p0;                       // lj_radius
        tp[tid][1] = 0.05f + 0.2f * p1;               // lj_wdepth
        tp[tid][2] = (-5.0f + 10.0f * p2) / lam;      // dgfree / lambda
        tp[tid][3] = 1.0f / lam;                      // 1 / lambda
        tp[tid][4] = 5.0f + 25.0f * p4;               // lk_volume
    }
    __syncthreads();

    // ---- phase 2: gather row-atom params for this tile ----
    if (tid < 16) {
        int t = atom_types[(size_t)p * N_ATOM + iBase + tid];
#pragma unroll
        for (int k = 0; k < 5; ++k) rowp[tid][k] = tp[t][k];
    }
    __syncthreads();

    // row params into registers (each lane's 8 rows: r + half*8, per C/D layout)
    float rr[8], rw[8], rdgl[8], rinvl[8], rvol[8];
#pragma unroll
    for (int r = 0; r < 8; ++r) {
        int m = r + half * 8;
        rr[r]    = rowp[m][0];
        rw[r]    = rowp[m][1];
        rdgl[r]  = rowp[m][2];
        rinvl[r] = rowp[m][3];
        rvol[r]  = rowp[m][4];
    }

    // ---- A fragment: row i = iBase+col, A[i,0..3] = (-2x, -2y, -2z, |c|^2) ----
    // 32-bit A 16x4 layout: V0 = K0 (lanes 0-15) / K2 (lanes 16-31); V1 = K1 / K3
    const size_t ca = ((size_t)p * N_ATOM + iBase + col) * 3;
    float ax = coords[ca + 0], ay = coords[ca + 1], az = coords[ca + 2];
    float sqi = ax * ax + ay * ay + az * az;
    v2f A;
    A[0] = half ? (-2.0f * az) : (-2.0f * ax);
    A[1] = half ? sqi          : (-2.0f * ay);

    const float X6L    = 21.433470507544f;   // (1/0.6)^6
    const float SLOPEK = 12.0f * (X6L - X6L * X6L);
    const float LKPREF = 0.08979353f;        // 1/(2*pi^1.5)

    float lj_sum = 0.0f, lk_sum = 0.0f;

    for (int tj = wave; tj < NT; tj += 8) {
        const int jBase = tj * 16;
        const int jg    = jBase + col;

        // B fragment: col j, B[0..3,j] = (x, y, z, 1); V0 = K0/K2, V1 = K1/K3
        const size_t cb = ((size_t)p * N_ATOM + jg) * 3;
        float bx = coords[cb + 0], by = coords[cb + 1], bz = coords[cb + 2];
        float sqj = bx * bx + by * by + bz * bz;
        v2f B;
        B[0] = half ? bz   : bx;
        B[1] = half ? 1.0f : by;

        // d2 tile via matrix pipe: D[i,j] = sq_i - 2*ci.cj  (EXEC all-1s here)
        v8f acc = {};
        acc = __builtin_amdgcn_wmma_f32_16x16x4_f32(
            false, A, false, B, (short)0, acc, false, false);

        // column-atom params (LDS broadcast)
        int tyj = atom_types[(size_t)p * N_ATOM + jg];
        float crj   = tp[tyj][0];
        float cwj   = tp[tyj][1];
        float cdglj = tp[tyj][2];
        float cinvj = tp[tyj][3];
        float cvolj = tp[tyj][4];

        // prefetch next j-tile's path_dist rows into cache
        if (tj + 8 < NT)
            __builtin_prefetch(
                &path_dist[((size_t)p * N_ATOM + iBase + col) * N_ATOM + jBase + 128],
                0, 1);

#pragma unroll
        for (int r = 0; r < 8; ++r) {
            const int ig = iBase + r + half * 8;    // C/D VGPR r -> row r / r+8
            float d2v = acc[r] + sqj;
            d2v = fmaxf(d2v, 1e-12f);
            float d = __builtin_amdgcn_sqrtf(d2v);

            // cubic fade: zero beyond 6.0 A -> skip memory + heavy math
            float t01  = fminf(fmaxf((d - 4.5f) * (1.0f / 1.5f), 0.0f), 1.0f);
            float fade = 1.0f - t01 * t01 * (3.0f - 2.0f * t01);

            if (fade > 0.0f && ig != jg) {
                int pd = __builtin_nontemporal_load(
                    &path_dist[((size_t)p * N_ATOM + ig) * N_ATOM + jg]);
                float cp  = (pd < 4) ? 0.0f : ((pd == 4) ? 0.2f : 1.0f);
                float wgt = cp * fade;
                if (wgt > 0.0f) {
                    // --- LJ 12-6 with linear extrapolation below 0.6*sigma ---
                    float sigma = rr[r] + crj;
                    float eps   = __builtin_amdgcn_sqrtf(rw[r] * cwj);
                    float dlin  = 0.6f * sigma;
                    float deff  = fmaxf(d, dlin);
                    float ratio = sigma * __builtin_amdgcn_rcpf(deff);
                    float x2    = ratio * ratio;
                    float x6    = x2 * x2 * x2;
                    float lj    = eps * (x6 * x6 - 2.0f * x6);
                    if (d < dlin) {
                        float slope = eps * SLOPEK * __builtin_amdgcn_rcpf(dlin);
                        lj += slope * (d - dlin);
                    }
                    // --- LK isotropic desolvation (symmetrized) ---
                    float dlk    = fmaxf(d, 1.0f);
                    float inv_d2 = __builtin_amdgcn_rcpf(dlk * dlk);
                    float xi = (dlk - rr[r]) * rinvl[r];
                    float xj = (dlk - crj)   * cinvj;
                    float exi = __expf(-xi * xi);
                    float exj = __expf(-xj * xj);
                    float lk = -LKPREF * inv_d2 *
                               (rdgl[r] * cvolj * exi + cdglj * rvol[r] * exj);

                    lj_sum += wgt * lj;
                    lk_sum += wgt * lk;
                }
            }
        }
    }

    lj_sum = wave_reduce_add(lj_sum);
    lk_sum = wave_reduce_add(lk_sum);
    if (lane == 0) {
        atomicAdd(&out[p],          0.5f * lj_sum);
        atomicAdd(&out[P_POSE + p], 0.5f * lk_sum);
    }
}

extern "C" void kernel_launch(void* const* d_in, const int* in_sizes, int n_in,
                              void* d_out, int out_size, void* d_ws, size_t ws_size,
                              hipStream_t stream) {
    const float* coords      = (const float*)d_in[0];
    const float* ljlk_params = (const float*)d_in[1];
    const int*   atom_types  = (const int*)d_in[2];
    const int*   path_dist   = (const int*)d_in[3];
    float* out = (float*)d_out;

    // harness poisons d_out; kernel accumulates with atomics -> zero first
    zero_out_kernel<<<1, 32, 0, stream>>>(out, out_size);
    ljlk_pose_kernel<<<P_POSE * NT, 256, 0, stream>>>(
        coords, ljlk_params, atom_types, path_dist, out);
}